// UberModel4_25598005084843
// MI455X (gfx1250) — compile-verified
//
#include <hip/hip_runtime.h>

// ---------------------------------------------------------------------------
// UberModel4: x(8M,5) streaming pointwise + 5x5-matvec chain.
// Memory floor: 320MB @ 23.3TB/s ~= 13.7us. Hot loop: scalar-per-lane VALU
// (constants in single SGPRs feeding v_fmac; v_cos/v_sin/v_log/v_tanh raw,
// all prescales folded into the matrices; NT b128 stores; global_prefetch).
// The one true matrix-matrix product (setup fusion M2 = Ar@Ar, 5x5 x 5x5)
// runs on the matrix unit via two chained V_WMMA_F32_16X16X4_F32, with a
// SALU cross-check fallback so an ISA-layout surprise cannot corrupt output.
// ---------------------------------------------------------------------------

typedef float v4f __attribute__((ext_vector_type(4)));
typedef float v2fw __attribute__((ext_vector_type(2)));
typedef float v8fw __attribute__((ext_vector_type(8)));

static constexpr double PI_D     = 3.14159265358979323846;
static constexpr double LN2_D    = 0.69314718055994530942;
static constexpr float  INV2PI_F = (float)(1.0 / (2.0 * PI_D));        // cos/sin prescale
static constexpr float  LN2_2PI  = (float)(LN2_D / (2.0 * PI_D));      // ln2 * 1/2pi
static constexpr float  PRE_A    = (float)(8.0 / (2.0 * PI_D));        // preamble: (8x+7)/2pi
static constexpr float  PRE_B    = (float)(7.0 / (2.0 * PI_D));
static constexpr float  KT       = (float)(24.0 * PI_D / LN2_D);       // exp2 arg: 12*x/ln2, x = 2pi*s
static constexpr float  K12PI    = (float)(12.0 * PI_D);               // tanh arg: 6*x, x = 2pi*s

__device__ __forceinline__ float final_tanh(float s) {
#if __has_builtin(__builtin_amdgcn_tanhf)
  return __builtin_amdgcn_tanhf(s * K12PI);                 // v_tanh_f32
#elif __has_builtin(__builtin_amdgcn_tanh_f32)
  return __builtin_amdgcn_tanh_f32(s * K12PI);
#else
  float e = __builtin_amdgcn_exp2f(s * KT);
  float r = __builtin_amdgcn_rcpf(e + 1.0f);
  return fmaf(-2.0f, r, 1.0f);
#endif
}

// Process one row of 5 values held in registers. s[] enters as raw x, leaves
// as tanh(6 * final).
__device__ __forceinline__ void process_row(
    float* __restrict__ s,
    const float (&A1)[5][5],   // W1^T / 2pi          (fc1 -> v_sin directly)
    const float (&A2)[5][5],   // W1^T * ln2 / 2pi    (log2 in, cos-arg out)
    const float (&M2)[5][5],   // (2*W2^T)^2          (two fc2 steps fused)
    const float (&Bs)[5],      // b1 / 2pi
    const float (&C2)[5])      // fused fc2 bias
{
#pragma unroll
  for (int i = 0; i < 5; ++i)            // preamble ((x+1)*2-1)^3 = 8x+7, prescaled
    s[i] = fmaf(s[i], PRE_A, PRE_B);

#pragma unroll
  for (int it = 0; it < 3; ++it) {
    float u[5], t[5], w[5];
#pragma unroll
    for (int i = 0; i < 5; ++i) u[i] = __builtin_amdgcn_cosf(s[i]);   // cos(2pi*s)
#pragma unroll
    for (int j = 0; j < 5; ++j) {                                     // fc1 (sin-prescaled)
      float a = Bs[j];
#pragma unroll
      for (int i = 0; i < 5; ++i) a = fmaf(u[i], A1[i][j], a);
      t[j] = a;
    }
#pragma unroll
    for (int i = 0; i < 5; ++i) t[i] = __builtin_amdgcn_sinf(t[i]);   // sin(2pi*t)
#pragma unroll
    for (int j = 0; j < 5; ++j) {                                     // fused double fc2
      float a = C2[j];
#pragma unroll
      for (int i = 0; i < 5; ++i) a = fmaf(t[i], M2[i][j], a);
      w[j] = a;
    }
#pragma unroll
    for (int i = 0; i < 5; ++i) w[i] = __builtin_amdgcn_logf(w[i]);   // log2; ln2 folded into A2
#pragma unroll
    for (int j = 0; j < 5; ++j) {                                     // fc1 (cos-prescaled out)
      float a = Bs[j];
#pragma unroll
      for (int i = 0; i < 5; ++i) a = fmaf(w[i], A2[i][j], a);
      s[j] = a;
    }
  }
#pragma unroll
  for (int i = 0; i < 5; ++i) s[i] = final_tanh(s[i]);
}

__global__ __launch_bounds__(256)
void ubermodel4_kernel(const float* __restrict__ x,
                       const float* __restrict__ W1,
                       const float* __restrict__ b1,
                       const float* __restrict__ W2,
                       const float* __restrict__ b2,
                       float* __restrict__ out,
                       long long batch)
{
  // ---- uniform per-thread constant prep (SALU float ops; amortized ~32 rows)
  float A1[5][5], A2[5][5], M2[5][5], Bs[5], C2[5];
  float Ar[5][5], d[5];
#pragma unroll
  for (int j = 0; j < 5; ++j) {
    float rs = 0.0f;
#pragma unroll
    for (int k = 0; k < 5; ++k) {
      float w = W2[j * 5 + k];
      Ar[k][j] = 2.0f * w;                // A = 2*W2^T  (row-vector convention)
      rs += w;
    }
    d[j] = b2[j] - 4.0f - 8.0f * rs;      // per-step bias
  }
#pragma unroll
  for (int k = 0; k < 5; ++k)
#pragma unroll
    for (int j = 0; j < 5; ++j) {
      float acc = 0.0f;
#pragma unroll
      for (int m = 0; m < 5; ++m) acc = fmaf(Ar[k][m], Ar[m][j], acc);
      M2[k][j] = acc;                     // M2 = Ar@Ar (SALU reference)
    }
#pragma unroll
  for (int j = 0; j < 5; ++j) {
    float acc = d[j];
#pragma unroll
    for (int k = 0; k < 5; ++k) acc = fmaf(d[k], Ar[k][j], acc);
    C2[j] = acc;                          // C2 = d@A + d
  }
#pragma unroll
  for (int i = 0; i < 5; ++i)
#pragma unroll
    for (int j = 0; j < 5; ++j) {
      float w = W1[j * 5 + i];            // W1^T
      A1[i][j] = w * INV2PI_F;
      A2[i][j] = w * LN2_2PI;
    }
#pragma unroll
  for (int j = 0; j < 5; ++j) Bs[j] = b1[j] * INV2PI_F;

#if __has_builtin(__builtin_amdgcn_wmma_f32_16x16x4_f32)
  // ---- M2 = Ar@Ar on the matrix unit: D(16x16) = A(16x4)xB(4x16) + C,
  // two chained K=4 steps (K=5 zero-padded). Uniform control flow here, so
  // EXEC is all-ones as WMMA requires. Ar staged through 128B of LDS to
  // build the striped operand layouts; invalid indices clamp to a zero slot.
  {
    __shared__ float lsm[32];
    if (threadIdx.x == 0) {
#pragma unroll
      for (int k = 0; k < 5; ++k)
#pragma unroll
        for (int j = 0; j < 5; ++j) lsm[k * 5 + j] = Ar[k][j];
#pragma unroll
      for (int i = 25; i < 32; ++i) lsm[i] = 0.0f;
    }
    __syncthreads();

    const int lane = (int)(threadIdx.x & 31u);   // wave32
    const int lo   = lane & 15;
    const bool hiH = lane >= 16;
    const bool rv  = lo < 5;
    // A (16x4): lanes 0-15 hold M=lane, K=0/1 in v0/v1; lanes 16-31 K=2/3.
    const float a0  = lsm[rv ? (hiH ? lo * 5 + 2 : lo * 5 + 0) : 30];
    const float a1  = lsm[rv ? (hiH ? lo * 5 + 3 : lo * 5 + 1) : 30];
    const float a20 = (hiH || !rv) ? 0.0f : lsm[lo * 5 + 4];
    // B (4x16): lanes 0-15 hold N=lane, K=0/1 in v0/v1; lanes 16-31 K=2/3.
    const float b0  = lsm[rv ? (hiH ? 10 + lo : 0 + lo) : 30];
    const float b1v = lsm[rv ? (hiH ? 15 + lo : 5 + lo) : 30];
    const float b20 = (hiH || !rv) ? 0.0f : lsm[20 + lo];

    v2fw Aop = {a0, a1}, Bop = {b0, b1v};
    v2fw Aop2 = {a20, 0.0f}, Bop2 = {b20, 0.0f};
    v8fw Cacc = {};
    Cacc = __builtin_amdgcn_wmma_f32_16x16x4_f32(
        false, Aop, false, Bop, (short)0, Cacc, false, false);
    Cacc = __builtin_amdgcn_wmma_f32_16x16x4_f32(
        false, Aop2, false, Bop2, (short)0, Cacc, false, false);

    // C/D: element (M=m,N=n), m<8,n<16 -> VGPR m, lane n. Broadcast back and
    // accept only if it matches the SALU reference (layout-safe fallback).
#pragma unroll
    for (int m = 0; m < 5; ++m)
#pragma unroll
      for (int n = 0; n < 5; ++n) {
        float wv  = __int_as_float(
            __builtin_amdgcn_readlane(__float_as_int(Cacc[m]), n));
        float ref = M2[m][n];
        M2[m][n]  = (fabsf(wv - ref) <= 1e-4f + 1e-3f * fabsf(ref)) ? wv : ref;
      }
  }
#endif

  const long long ngroups = batch >> 2;                    // 4 rows (80B) per thread
  const long long stride  = (long long)gridDim.x * blockDim.x;
  for (long long g = (long long)blockIdx.x * blockDim.x + threadIdx.x;
       g < ngroups; g += stride) {
    // gfx1250 path: prefetch next grid-stride tile (global_prefetch_b8, speculative).
    if (g + stride < ngroups)
      __builtin_prefetch(x + (g + stride) * 20, 0, 0);

    const v4f* src = reinterpret_cast<const v4f*>(x + g * 20);
    v4f p0 = src[0], p1 = src[1], p2 = src[2], p3 = src[3], p4 = src[4];
    float v[20] = {p0.x, p0.y, p0.z, p0.w,
                   p1.x, p1.y, p1.z, p1.w,
                   p2.x, p2.y, p2.z, p2.w,
                   p3.x, p3.y, p3.z, p3.w,
                   p4.x, p4.y, p4.z, p4.w};
#pragma unroll
    for (int r = 0; r < 4; ++r)
      process_row(&v[r * 5], A1, A2, M2, Bs, C2);

    // Write-once output stream: non-temporal (TH_NT) b128 stores.
    v4f* dst = reinterpret_cast<v4f*>(out + g * 20);
    v4f q0 = {v[0],  v[1],  v[2],  v[3]};
    v4f q1 = {v[4],  v[5],  v[6],  v[7]};
    v4f q2 = {v[8],  v[9],  v[10], v[11]};
    v4f q3 = {v[12], v[13], v[14], v[15]};
    v4f q4 = {v[16], v[17], v[18], v[19]};
    __builtin_nontemporal_store(q0, dst + 0);
    __builtin_nontemporal_store(q1, dst + 1);
    __builtin_nontemporal_store(q2, dst + 2);
    __builtin_nontemporal_store(q3, dst + 3);
    __builtin_nontemporal_store(q4, dst + 4);
  }

  // ---- scalar tail for batch % 4 (BATCH=8388608 has none, but stay generic)
  const long long tail = batch & 3LL;
  if (__builtin_expect(tail != 0, 0) && blockIdx.x == 0 && threadIdx.x < (unsigned)tail) {
    const long long r = (batch & ~3LL) + threadIdx.x;
    float s[5];
#pragma unroll
    for (int i = 0; i < 5; ++i) s[i] = x[r * 5 + i];
    process_row(s, A1, A2, M2, Bs, C2);
#pragma unroll
    for (int i = 0; i < 5; ++i) out[r * 5 + i] = s[i];
  }
}

extern "C" void kernel_launch(void* const* d_in, const int* in_sizes, int n_in,
                              void* d_out, int out_size, void* d_ws, size_t ws_size,
                              hipStream_t stream) {
  const float* x  = (const float*)d_in[0];
  const float* W1 = (const float*)d_in[1];
  const float* b1 = (const float*)d_in[2];
  const float* W2 = (const float*)d_in[3];
  const float* b2 = (const float*)d_in[4];
  float* out = (float*)d_out;

  const long long batch  = (long long)in_sizes[0] / 5;
  const long long groups = batch >> 2;
  const int threads = 256;
  long long want = (groups + threads - 1) / threads;
  if (want < 1) want = 1;
  const int blocks = (int)(want < 1024 ? want : 1024);   // ~8K waves, 32 rows/thread

  ubermodel4_kernel<<<blocks, threads, 0, stream>>>(x, W1, b1, W2, b2, out, batch);
}